// Q0DistMultiTask_65326452572832
// MI455X (gfx1250) — compile-verified
//
#include <hip/hip_runtime.h>
#include <hip/hip_bf16.h>
#include <math.h>

// ---------------------------------------------------------------------------
// MI455X (gfx1250) implementation.
//
// Roofline: cov output = 8 * 3072*3072 * 4B = 302 MB -> ~13us at 23.3 TB/s.
// Everything else is noise, so: exploit Kronecker structure to avoid the
// 3072^3 W@W^T GEMM (58 GFLOP -> ~0.3 GFLOP), keep f32 precision, and use
// V_WMMA_F32_16X16X4_F32 for the one surviving dense GEMM stage (M_g Grams).
// ---------------------------------------------------------------------------

typedef __attribute__((ext_vector_type(2))) float v2f;
typedef __attribute__((ext_vector_type(8))) float v8f;

#define NT 256

// Problem constants (FREQ=24, GAMMA=1, ISO=1e-4, TASK_NOISE=1, shapes fixed).
#define SIGMA2 1.0001f
#define COVN   9437184   // 3072*3072
#define MEAN_N 24576     // 8*24*128

// Workspace layout (float offsets, all 64-aligned; total ~1.8 MB)
enum : int {
  OFF_KC   = 0,      // 24x24
  OFF_KFC  = 640,    // 24x24
  OFF_KFF  = 1280,   // 24x24
  OFF_BT   = 1920,   // 128x128
  OFF_P    = 18304,  // 128x128 eigenvectors of Bt
  OFF_LAM  = 34688,  // 128
  OFF_U    = 34816,  // 24x24 eigenvectors of Kc
  OFF_S    = 35392,  // 24
  OFF_A    = 35456,  // 24x24  A = Kfc @ U
  OFF_CM   = 36032,  // 128x128 Cm = Bt @ P
  OFF_INVD = 52416,  // 24x128  1/(s_g*lam_m + sigma2)
  OFF_M    = 55488,  // 24 x 128x128  M_g Gram matrices
};

__device__ inline void rr_pair(int r, int k, int n, int& p, int& q) {
  int m = n - 1;
  int a, b;
  if (k == 0) { a = m; b = r % m; }
  else { a = (r + k) % m; b = (r - k + m) % m; }
  p = a < b ? a : b;
  q = a < b ? b : a;
}

// Block-parallel cyclic Jacobi eigensolver (round-robin disjoint rotations).
// A, Vv in LDS (n*n each). On exit: diag(A)=eigenvalues, Vv columns=eigvecs.
__device__ void jacobi_eigh(float* A, float* Vv, int n, int sweeps,
                            int tid, float* cbuf, float* sbuf) {
  int np = n >> 1;
  for (int i = tid; i < n * n; i += NT) Vv[i] = ((i / n) == (i % n)) ? 1.f : 0.f;
  __syncthreads();
  for (int sw = 0; sw < sweeps; ++sw) {
    for (int r = 0; r < n - 1; ++r) {
      if (tid < np) {
        int p, q; rr_pair(r, tid, n, p, q);
        float apq = A[p * n + q];
        float c = 1.f, s = 0.f;
        if (fabsf(apq) > 1e-20f) {
          float tau = (A[q * n + q] - A[p * n + p]) / (2.f * apq);
          float t = (tau >= 0.f) ? 1.f / (tau + sqrtf(1.f + tau * tau))
                                 : 1.f / (tau - sqrtf(1.f + tau * tau));
          c = 1.f / sqrtf(1.f + t * t);
          s = t * c;
        }
        cbuf[tid] = c; sbuf[tid] = s;
      }
      __syncthreads();
      // Row rotations: A <- J^T A   (disjoint row pairs)
      for (int idx = tid; idx < np * n; idx += NT) {
        int k = idx / n, col = idx - k * n;
        int p, q; rr_pair(r, k, n, p, q);
        float c = cbuf[k], s = sbuf[k];
        float a1 = A[p * n + col], a2 = A[q * n + col];
        A[p * n + col] = c * a1 - s * a2;
        A[q * n + col] = s * a1 + c * a2;
      }
      __syncthreads();
      // Column rotations: A <- A J ; accumulate eigenvectors V <- V J
      for (int idx = tid; idx < np * n; idx += NT) {
        int k = idx / n, row = idx - k * n;
        int p, q; rr_pair(r, k, n, p, q);
        float c = cbuf[k], s = sbuf[k];
        float a1 = A[row * n + p], a2 = A[row * n + q];
        A[row * n + p] = c * a1 - s * a2;
        A[row * n + q] = s * a1 + c * a2;
        float v1 = Vv[row * n + p], v2 = Vv[row * n + q];
        Vv[row * n + p] = c * v1 - s * v2;
        Vv[row * n + q] = s * v1 + c * v2;
      }
      __syncthreads();
    }
  }
}

// Sort eigenpairs ascending (LAPACK-style) + deterministic sign fix.
__device__ void sort_sign(float* A, float* Vv, int n, int tid, float* sc) {
  for (int i = 0; i < n - 1; ++i) {
    if (tid == 0) {
      int mi = i; float mv = A[i * n + i];
      for (int j = i + 1; j < n; ++j) {
        float v = A[j * n + j];
        if (v < mv) { mv = v; mi = j; }
      }
      sc[0] = __int_as_float(mi);
    }
    __syncthreads();
    int mi = __float_as_int(sc[0]);
    if (mi != i) {
      if (tid == 0) {
        float t = A[i * n + i]; A[i * n + i] = A[mi * n + mi]; A[mi * n + mi] = t;
      }
      for (int k = tid; k < n; k += NT) {
        float t = Vv[k * n + i]; Vv[k * n + i] = Vv[k * n + mi]; Vv[k * n + mi] = t;
      }
    }
    __syncthreads();
  }
  for (int j = tid; j < n; j += NT) {   // make max-|.| component positive
    float mv = -1.f; int mk = 0;
    for (int k = 0; k < n; ++k) {
      float av = fabsf(Vv[k * n + j]);
      if (av > mv) { mv = av; mk = k; }
    }
    if (Vv[mk * n + j] < 0.f)
      for (int k = 0; k < n; ++k) Vv[k * n + j] = -Vv[k * n + j];
  }
  __syncthreads();
}

// ---------------------------------------------------------------------------
// Kernel 1: one workgroup. Kernel matrices, Bt, both eigh's, Cm, A, 1/D.
// Uses 128.6 KB dynamic LDS (legal: CDNA5 WGP has 320 KB).
// ---------------------------------------------------------------------------
__global__ __launch_bounds__(NT) void setup_kernel(const float* __restrict__ Vmat,
                                                   const float* __restrict__ task_var,
                                                   float* ws) {
  extern __shared__ float smem[];
  float* shA  = smem;            // 16384
  float* shV  = smem + 16384;    // 16384
  float* cbuf = smem + 32768;    // 64
  float* sbuf = smem + 32832;    // 64
  float* sc   = smem + 32896;    // 8
  int tid = threadIdx.x;
  const float dt = 3.14159265358979323846f / 24.f;

  // Matern-1/2 kernel matrices (ell=1): Kc, Kff over ctx/fut grid, Kfc cross.
  for (int idx = tid; idx < 576; idx += NT) {
    int i = idx / 24, j = idx % 24;
    float v = __expf(-fabsf((float)(i - j)) * dt);
    ws[OFF_KC + idx]  = v;
    ws[OFF_KFF + idx] = v;
    ws[OFF_KFC + idx] = __expf(-(float)(i + 24 - j) * dt);   // always > 0
  }
  // Bt = V V^T + diag(task_var)
  for (int idx = tid; idx < 16384; idx += NT) {
    int c = idx >> 7, d = idx & 127;
    float acc = 0.f;
    for (int r = 0; r < 8; ++r) acc += Vmat[c * 8 + r] * Vmat[d * 8 + r];
    if (c == d) acc += task_var[c];
    shA[idx] = acc;
    ws[OFF_BT + idx] = acc;
  }
  __syncthreads();

  // eigh(Bt): 12 sweeps of parallel Jacobi (n=128)
  jacobi_eigh(shA, shV, 128, 12, tid, cbuf, sbuf);
  sort_sign(shA, shV, 128, tid, sc);
  for (int idx = tid; idx < 16384; idx += NT) ws[OFF_P + idx] = shV[idx];
  for (int i = tid; i < 128; i += NT) ws[OFF_LAM + i] = shA[i * 128 + i];
  __syncthreads();

  // Cm = Bt @ P  (P still live in shV)
  for (int idx = tid; idx < 16384; idx += NT) {
    int c = idx >> 7, m = idx & 127;
    float acc = 0.f;
    for (int k = 0; k < 128; ++k) acc += ws[OFF_BT + c * 128 + k] * shV[k * 128 + m];
    ws[OFF_CM + idx] = acc;
  }
  __syncthreads();

  // eigh(Kc): n=24
  for (int idx = tid; idx < 576; idx += NT) shA[idx] = ws[OFF_KC + idx];
  __syncthreads();
  jacobi_eigh(shA, shV, 24, 16, tid, cbuf, sbuf);
  sort_sign(shA, shV, 24, tid, sc);
  for (int idx = tid; idx < 576; idx += NT) ws[OFF_U + idx] = shV[idx];
  for (int i = tid; i < 24; i += NT) ws[OFF_S + i] = shA[i * 24 + i];
  __syncthreads();

  // A = Kfc @ U  (U live in shV)
  for (int idx = tid; idx < 576; idx += NT) {
    int f = idx / 24, g = idx % 24;
    float acc = 0.f;
    for (int k = 0; k < 24; ++k) acc += ws[OFF_KFC + f * 24 + k] * shV[k * 24 + g];
    ws[OFF_A + idx] = acc;
  }
  __syncthreads();
  // invD[g][m] = 1 / (s_g * lam_m + sigma2)
  for (int idx = tid; idx < 3072; idx += NT) {
    int g = idx >> 7, m = idx & 127;
    ws[OFF_INVD + idx] = 1.f / (ws[OFF_S + g] * ws[OFF_LAM + m] + SIGMA2);
  }
}

// ---------------------------------------------------------------------------
// Kernel 2: M_g = (Cm * invD_g) @ Cm^T via V_WMMA_F32_16X16X4_F32.
// One wave per 16x16 output tile; 32 WMMA issues per tile (K=128, step 4).
// A-frag layout (f32 16x4): lane L<16 -> row L, K={k0,k0+1}; L>=16 -> K={k0+2,k0+3}.
// C/D layout: VGPR v -> M = v + 8*(L>=16), N = L&15.
// ---------------------------------------------------------------------------
__global__ __launch_bounds__(32) void mg_kernel(float* ws) {
  const float* Cm   = ws + OFF_CM;
  const float* invD = ws + OFF_INVD + blockIdx.z * 128;
  const int ci = blockIdx.y, cj = blockIdx.x;
  const int lane = threadIdx.x;
  const int half = lane >> 4;
  const int l15  = lane & 15;
  const int kb   = half * 2;
  const int arow = ci * 16 + l15;      // A rows (M index)
  const int brow = cj * 16 + l15;      // B cols (N index): B[k][n] = Cm[n][k]
  v8f acc = {};
#pragma unroll
  for (int k0 = 0; k0 < 128; k0 += 4) {
    int k = k0 + kb;
    v2f a, b;
    a.x = Cm[arow * 128 + k]     * invD[k];
    a.y = Cm[arow * 128 + k + 1] * invD[k + 1];
    b.x = Cm[brow * 128 + k];
    b.y = Cm[brow * 128 + k + 1];
    acc = __builtin_amdgcn_wmma_f32_16x16x4_f32(false, a, false, b,
                                                (short)0, acc, false, false);
  }
  float* Mg = ws + OFF_M + blockIdx.z * 16384;
#pragma unroll
  for (int v = 0; v < 8; ++v) {
    int m = ci * 16 + v + 8 * half;
    Mg[m * 128 + cj * 16 + l15] = acc[v];
  }
}

// ---------------------------------------------------------------------------
// Kernel 3: mean, one block per batch. Replicates the reference's einsum chain
// exactly: alpha = U((U Y P)/D)P^T ; mean = Kfc alpha Bt + fut_loc.
// ---------------------------------------------------------------------------
__global__ __launch_bounds__(NT) void mean_kernel(const float* __restrict__ x,
                                                  const float* __restrict__ ws,
                                                  float* __restrict__ out) {
  int b = blockIdx.x, tid = threadIdx.x;
  __shared__ float shU[576], shK[576];
  __shared__ float s1[3072], s2[3072], s3[3072];
  for (int i = tid; i < 576; i += NT) { shU[i] = ws[OFF_U + i]; shK[i] = ws[OFF_KFC + i]; }
  const float* xb = x + b * 128 * 48;
  // Y[l][c] = 0.5*(x[b,c,24+l] - x[b,c,l])
  for (int i = tid; i < 3072; i += NT) {
    int l = i >> 7, c = i & 127;
    s1[i] = 0.5f * (xb[c * 48 + 24 + l] - xb[c * 48 + l]);
  }
  __syncthreads();
  for (int i = tid; i < 3072; i += NT) {           // T1 = U @ Y
    int r = i >> 7, c = i & 127; float a = 0.f;
    for (int l = 0; l < 24; ++l) a += shU[r * 24 + l] * s1[l * 128 + c];
    s2[i] = a;
  }
  __syncthreads();
  for (int i = tid; i < 3072; i += NT) {           // Z = (T1 @ P) * invD
    int r = i >> 7, m = i & 127; float a = 0.f;
    for (int c = 0; c < 128; ++c) a += s2[r * 128 + c] * ws[OFF_P + c * 128 + m];
    s3[i] = a * ws[OFF_INVD + r * 128 + m];
  }
  __syncthreads();
  for (int i = tid; i < 3072; i += NT) {           // T3 = Z @ P^T
    int r = i >> 7, c2 = i & 127; float a = 0.f;
    for (int m = 0; m < 128; ++m) a += s3[r * 128 + m] * ws[OFF_P + c2 * 128 + m];
    s1[i] = a;
  }
  __syncthreads();
  for (int i = tid; i < 3072; i += NT) {           // alpha = U @ T3
    int r = i >> 7, c = i & 127; float a = 0.f;
    for (int l = 0; l < 24; ++l) a += shU[r * 24 + l] * s1[l * 128 + c];
    s2[i] = a;
  }
  __syncthreads();
  for (int i = tid; i < 3072; i += NT) {           // T5 = Kfc @ alpha
    int f = i >> 7, c = i & 127; float a = 0.f;
    for (int l = 0; l < 24; ++l) a += shK[f * 24 + l] * s2[l * 128 + c];
    s3[i] = a;
  }
  __syncthreads();
  for (int i = tid; i < 3072; i += NT) {           // mean = T5 @ Bt + loc_phase
    int f = i >> 7, d = i & 127; float a = 0.f;
    for (int c = 0; c < 128; ++c) a += s3[f * 128 + c] * ws[OFF_BT + c * 128 + d];
    a += 0.5f * (xb[d * 48 + f] + xb[d * 48 + 24 + f]);
    out[(b * 24 + f) * 128 + d] = a;
  }
}

// ---------------------------------------------------------------------------
// Kernel 4: cov assembly + 8-way broadcast (the 302 MB bandwidth-bound stage).
// cov[(f,c),(f',c')] = Kff[f,f']*Bt[c,c'] - sum_g A[f,g]A[f',g]*M_g[c,c'] + s2*I
// One block per (f,f') 128x128 tile; 25 FMA per element, float4 stores.
// ---------------------------------------------------------------------------
__global__ __launch_bounds__(NT) void cov_kernel(const float* __restrict__ ws,
                                                 float* __restrict__ out) {
  int f = blockIdx.y, f2 = blockIdx.x, tid = threadIdx.x;
  __shared__ float w[24];
  __shared__ float kffs;
  if (tid < 24) w[tid] = ws[OFF_A + f * 24 + tid] * ws[OFF_A + f2 * 24 + tid];
  if (tid == 0) kffs = ws[OFF_KFF + f * 24 + f2];
  __syncthreads();
  float kff = kffs;
  const float4* Bt4 = (const float4*)(ws + OFF_BT);
  const float4* M4  = (const float4*)(ws + OFF_M);
  float* covb = out + MEAN_N;
  for (int qi = tid; qi < 4096; qi += NT) {
    float4 acc = Bt4[qi];
    acc.x *= kff; acc.y *= kff; acc.z *= kff; acc.w *= kff;
#pragma unroll
    for (int g = 0; g < 24; ++g) {
      float wg = w[g];
      float4 m = M4[g * 4096 + qi];
      acc.x -= wg * m.x; acc.y -= wg * m.y; acc.z -= wg * m.z; acc.w -= wg * m.w;
    }
    int c = qi >> 5;            // row within tile
    int c2q = (qi & 31) * 4;    // first col of quad
    if (f == f2 && c >= c2q && c < c2q + 4)
      ((float*)&acc)[c - c2q] += SIGMA2;
    long row = (long)(f * 128 + c) * 3072 + f2 * 128 + c2q;
#pragma unroll
    for (int b = 0; b < 8; ++b)
      *(float4*)(covb + (long)b * COVN + row) = acc;
  }
}

// ---------------------------------------------------------------------------
extern "C" void kernel_launch(void* const* d_in, const int* in_sizes, int n_in,
                              void* d_out, int out_size, void* d_ws, size_t ws_size,
                              hipStream_t stream) {
  (void)in_sizes; (void)n_in; (void)out_size; (void)ws_size;
  const float* x  = (const float*)d_in[0];
  const float* V  = (const float*)d_in[1];
  const float* tv = (const float*)d_in[2];
  float* out = (float*)d_out;
  float* ws  = (float*)d_ws;

  size_t smemBytes = (size_t)(16384 + 16384 + 64 + 64 + 8) * sizeof(float); // 128.6 KB
  setup_kernel<<<1, NT, smemBytes, stream>>>(V, tv, ws);
  mg_kernel<<<dim3(8, 8, 24), 32, 0, stream>>>(ws);
  mean_kernel<<<8, NT, 0, stream>>>(x, ws, out);
  cov_kernel<<<dim3(24, 24), NT, 0, stream>>>(ws, out);
}